// AttentionUnroll_24455543783996
// MI455X (gfx1250) — compile-verified
//
#include <hip/hip_runtime.h>
#include <math.h>

// Problem constants (from the reference)
#define NH   128   // heads
#define HPD  256   // size per head
#define NE   (NH * HPD)         // 32768 floats per (H, D) tensor
#define NSTEPS 32               // SEQ_LEN - START_LEN

typedef float v2f __attribute__((ext_vector_type(2)));
typedef float v8f __attribute__((ext_vector_type(8)));

// ---------------------------------------------------------------------------
// Per-head GEMV: dst[h, :] = src[h, :] (1x256) @ W[h, :, :] (256x256)
// 256 threads; thread t owns 4 output columns (cg..cg+3) on row-group rg,
// so each wave issues contiguous 512B global_load_b128 from the weight row.
// Weights are streamed once (no reuse within a step) -> straight-to-VGPR loads,
// L2-resident after step 0 (134MB total weights < 192MB L2).
// ---------------------------------------------------------------------------
__device__ __forceinline__ void gemv_head(const float* __restrict__ src,
                                          const float* __restrict__ W,
                                          float* __restrict__ dst,
                                          int h)
{
    __shared__ float xs[HPD];          // the 256-float activation row
    __shared__ float part[4 * HPD];    // 4 row-group partials
    const int tid = threadIdx.x;

    xs[tid] = src[h * HPD + tid];
    __syncthreads();

    const float* Wh = W + (size_t)h * HPD * HPD;
    const int cg = (tid & 63) * 4;     // column base (0..252 step 4)
    const int rg = tid >> 6;           // row group 0..3

    float a0 = 0.f, a1 = 0.f, a2 = 0.f, a3 = 0.f;
    for (int d = rg; d < HPD; d += 4) {
        const float4 w = *(const float4*)(Wh + (size_t)d * HPD + cg);
        __builtin_prefetch(Wh + (size_t)(d + 8) * HPD + cg, 0, 0);
        const float xv = xs[d];
        a0 = fmaf(xv, w.x, a0);
        a1 = fmaf(xv, w.y, a1);
        a2 = fmaf(xv, w.z, a2);
        a3 = fmaf(xv, w.w, a3);
    }

    part[rg * HPD + cg + 0] = a0;
    part[rg * HPD + cg + 1] = a1;
    part[rg * HPD + cg + 2] = a2;
    part[rg * HPD + cg + 3] = a3;
    __syncthreads();

    const float r = part[tid] + part[HPD + tid] + part[2 * HPD + tid] + part[3 * HPD + tid];
    dst[h * HPD + tid] = r;
}

// Projections: grid (128, 3); blockIdx.y selects Wq/Wk/Wv -> q/kk/vv
__global__ void __launch_bounds__(256)
proj_gemv_kernel(const float* __restrict__ x,
                 const float* __restrict__ Wq,
                 const float* __restrict__ Wk,
                 const float* __restrict__ Wv,
                 float* __restrict__ q,
                 float* __restrict__ kk,
                 float* __restrict__ vv)
{
    const int h = blockIdx.x;
    const int which = blockIdx.y;
    const float* W = (which == 0) ? Wq : (which == 1) ? Wk : Wv;
    float* dst     = (which == 0) ? q  : (which == 1) ? kk : vv;
    gemv_head(x, W, dst, h);
}

// Output projection: grid (128)
__global__ void __launch_bounds__(256)
out_gemv_kernel(const float* __restrict__ y,
                const float* __restrict__ Wo,
                float* __restrict__ dst)
{
    gemv_head(y, Wo, dst, blockIdx.x);
}

// ---------------------------------------------------------------------------
// Attention core (fp32 WMMA 16x16x4):
//   S = 0.125 * kk(128x256) @ q^T(256x128);  P = softmax_rows(S);
//   y = P(128x128) @ vv(128x256)
// One block (8 waves) per 16-row slab of S / y  -> grid of 8 blocks.
// Phase 1: wave w computes the 16x16 S tile at columns j0 = 16*w (K = 256).
// Phase 2: softmax of the 16 rows in LDS.
// Phase 3: wave w computes y tiles at columns 32*w and 32*w+16 (K = 128).
// A fragment (16x4 f32): lane L<16 holds A[L][k0..k0+1], lane L+16 -> k0+2..3.
// B fragment (4x16 f32): lane n<16 holds B[k0..k0+1][n], lane n+16 -> k0+2..3.
// D (16x16 f32, 8 VGPRs): VGPR r, lanes 0-15 -> (M=r, N=lane); 16-31 -> (M=r+8).
// ---------------------------------------------------------------------------
__global__ void __launch_bounds__(256)
attn_wmma_kernel(const float* __restrict__ q,
                 const float* __restrict__ kk,
                 const float* __restrict__ vv,
                 float* __restrict__ y)
{
    __shared__ float S[16 * NH];       // 16 rows x 128 cols, 8 KB

    const int tid  = threadIdx.x;
    const int lane = tid & 31;
    const int wv   = tid >> 5;          // wave 0..7
    const int i0   = blockIdx.x * 16;   // row slab base

    const int mrow = lane & 15;         // M (or N) index within tile
    const int koff = (lane >> 4) * 2;   // K sub-offset per lane half
    const int mb   = (lane >> 4) * 8;   // D-fragment row base per lane half

    // ---- Phase 1: S tile (rows i0..i0+15, cols j0..j0+15), K = 256 ----
    {
        const int j0 = wv * 16;
        const float* arow = kk + (size_t)(i0 + mrow) * HPD + koff;  // A: kk rows
        const float* brow = q  + (size_t)(j0 + mrow) * HPD + koff;  // B: q rows (as q^T cols)
        v8f acc = {0.f, 0.f, 0.f, 0.f, 0.f, 0.f, 0.f, 0.f};
        #pragma unroll 8
        for (int k0 = 0; k0 < HPD; k0 += 4) {
            const v2f a = *(const v2f*)(arow + k0);
            const v2f b = *(const v2f*)(brow + k0);
            acc = __builtin_amdgcn_wmma_f32_16x16x4_f32(
                      false, a, false, b, (short)0, acc, false, false);
        }
        #pragma unroll
        for (int r = 0; r < 8; ++r)
            S[(mb + r) * NH + j0 + mrow] = acc[r] * 0.125f;
    }
    __syncthreads();

    // ---- Phase 2: row softmax over 128 columns (threads 0..15, no WMMA) ----
    if (tid < 16) {
        float* row = S + tid * NH;
        float m = -3.402823466e38f;
        for (int j = 0; j < NH; ++j) m = fmaxf(m, row[j]);
        float s = 0.f;
        for (int j = 0; j < NH; ++j) { const float e = __expf(row[j] - m); row[j] = e; s += e; }
        const float inv = 1.f / s;
        for (int j = 0; j < NH; ++j) row[j] *= inv;
    }
    __syncthreads();

    // ---- Phase 3: y = P @ vv ; wave w owns output columns [32w, 32w+32) ----
    #pragma unroll
    for (int t = 0; t < 2; ++t) {
        const int n0   = wv * 32 + t * 16;
        const int ncol = n0 + mrow;
        v8f acc = {0.f, 0.f, 0.f, 0.f, 0.f, 0.f, 0.f, 0.f};
        #pragma unroll 4
        for (int k0 = 0; k0 < NH; k0 += 4) {
            const v2f a = { S[mrow * NH + k0 + koff],
                            S[mrow * NH + k0 + koff + 1] };            // P rows (LDS)
            const v2f b = { vv[(size_t)(k0 + koff)     * HPD + ncol],
                            vv[(size_t)(k0 + koff + 1) * HPD + ncol] };// vv rows = K dim
            acc = __builtin_amdgcn_wmma_f32_16x16x4_f32(
                      false, a, false, b, (short)0, acc, false, false);
        }
        #pragma unroll
        for (int r = 0; r < 8; ++r)
            y[(size_t)(i0 + mb + r) * HPD + ncol] = acc[r];
    }
}

// ---------------------------------------------------------------------------
// Launch: 32 dependent steps, 3 kernels each, ping-ponging x through d_ws.
// d_in order: x, k(dead), v(dead), weight_q, weight_k, weight_v, weight_o
// ---------------------------------------------------------------------------
extern "C" void kernel_launch(void* const* d_in, const int* in_sizes, int n_in,
                              void* d_out, int out_size, void* d_ws, size_t ws_size,
                              hipStream_t stream)
{
    (void)in_sizes; (void)n_in; (void)out_size; (void)ws_size;

    const float* x  = (const float*)d_in[0];
    const float* Wq = (const float*)d_in[3];
    const float* Wk = (const float*)d_in[4];
    const float* Wv = (const float*)d_in[5];
    const float* Wo = (const float*)d_in[6];
    float* out = (float*)d_out;
    float* ws  = (float*)d_ws;

    float* xb0 = ws;                 // x ping
    float* xb1 = ws + 1 * NE;        // x pong
    float* qb  = ws + 2 * NE;
    float* kkb = ws + 3 * NE;
    float* vvb = ws + 4 * NE;
    float* yb  = ws + 5 * NE;        // total 768 KB of workspace

    hipMemcpyAsync(xb0, x, (size_t)NE * sizeof(float),
                   hipMemcpyDeviceToDevice, stream);

    for (int s = 0; s < NSTEPS; ++s) {
        const float* xc = (s & 1) ? xb1 : xb0;
        float*       xn = (s == NSTEPS - 1) ? out : ((s & 1) ? xb0 : xb1);

        proj_gemv_kernel<<<dim3(NH, 3), 256, 0, stream>>>(xc, Wq, Wk, Wv, qb, kkb, vvb);
        attn_wmma_kernel<<<dim3(NH / 16), 256, 0, stream>>>(qb, kkb, vvb, yb);
        out_gemv_kernel<<<dim3(NH), 256, 0, stream>>>(yb, Wo, xn);
    }
}